// RAGRetrievalLoss_84688165143168
// MI455X (gfx1250) — compile-verified
//
#include <hip/hip_runtime.h>

// RAG retrieval hinge loss on MI455X (gfx1250, wave32).
//
// hinge[b,i,j] = relu(M - s_i + s_j) * rel_i * irr_j
// Since masks are {0,1}: = relu( (M - s_i + s_j) * rel_i * irr_j )
// and the pre-relu masked matrix is rank-2:
//   rel_i * (irr_j*(M+s_j))  +  (-rel_i*s_i) * irr_j
// => each 16x16 tile is one exact-fp32 V_WMMA_F32_16X16X4_F32 (K=2 used).
// relu+reduce uses relu(x) = (x + |x|)/2 so |x| folds into a v_add source
// modifier (1 VALU op) instead of fmaxf's 2x v_max_num_f32 lowering.

typedef __attribute__((ext_vector_type(2))) float v2f;
typedef __attribute__((ext_vector_type(8))) float v8f;

#define RAG_MARGIN 0.5f
#define RAG_BATCH 2048
#define RAG_DOCS 128
#define WAVES_PER_BLOCK 8

__global__ void rag_zero_ws(float* ws) {
    if (threadIdx.x < 2) ws[threadIdx.x] = 0.0f;
}

__global__ __launch_bounds__(WAVES_PER_BLOCK * 32)
void rag_loss_kernel(const float* __restrict__ scores,
                     const int* __restrict__ relevance,
                     float* __restrict__ ws) {
    const int lane = threadIdx.x & 31;
    const int wave = threadIdx.x >> 5;
    const int b    = blockIdx.x * WAVES_PER_BLOCK + wave;   // one wave per batch row

    const float* s = scores    + b * RAG_DOCS;
    const int*   r = relevance + b * RAG_DOCS;

    // Per-lane operand slices. Row/col index = t*16 + (lane & 15).
    // Lanes 16..31 are forced to zero: they hold the unused K=2/K=3 rows of
    // the 16x4 / 4x16 fp32 WMMA operands (ISA 7.12.2 layout).
    const bool lo = (lane < 16);
    const int  li = lane & 15;

    float aRel[8];   // A, K=0 row: rel_i
    float aNRS[8];   // A, K=1 row: -rel_i * s_i
    float b0v[8];    // B, K=0 row: irr_j * (M + s_j)
    float b1v[8];    // B, K=1 row: irr_j
#pragma unroll
    for (int t = 0; t < 8; ++t) {
        float sv = s[t * 16 + li];
        int   rv = r[t * 16 + li];
        float isRel = (lo && rv != 0) ? 1.0f : 0.0f;
        float isIrr = (lo && rv == 0) ? 1.0f : 0.0f;
        aRel[t] = isRel;
        aNRS[t] = -isRel * sv;
        b0v[t]  = isIrr * (RAG_MARGIN + sv);
        b1v[t]  = isIrr;
    }

    // Pair count for this batch: nrel * nirr  (== pair_mask.sum() per batch).
    float nrel = 0.0f, nirr = 0.0f;
#pragma unroll
    for (int t = 0; t < 8; ++t) { nrel += aRel[t]; nirr += b1v[t]; }
#pragma unroll
    for (int off = 16; off > 0; off >>= 1) {
        nrel += __shfl_down(nrel, off, 32);
        nirr += __shfl_down(nirr, off, 32);
    }

    // 8x8 grid of 16x16 tiles; each tile = one fp32 WMMA, then relu+reduce
    // the 8 accumulator VGPRs via acc += d + |d| (halved at the end).
    // EXEC is uniform (no divergent control flow), as WMMA requires.
    v8f accv = {};
#pragma unroll
    for (int ti = 0; ti < 8; ++ti) {
        v2f a;
        a.x = aRel[ti];   // K=0 (lanes 0-15); lanes 16-31 already 0 (K=2)
        a.y = aNRS[ti];   // K=1 (lanes 0-15); lanes 16-31 already 0 (K=3)
#pragma unroll
        for (int tj = 0; tj < 8; ++tj) {
            v2f bb;
            bb.x = b0v[tj];
            bb.y = b1v[tj];
            v8f c = {};
            v8f d = __builtin_amdgcn_wmma_f32_16x16x4_f32(
                /*neg_a=*/false, a, /*neg_b=*/false, bb,
                /*c_mod=*/(short)0, c, /*reuse_a=*/false, /*reuse_b=*/false);
#pragma unroll
            for (int k = 0; k < 8; ++k)
                accv[k] += d[k] + __builtin_fabsf(d[k]);   // 2*relu(d[k])
        }
    }

    float total = 0.0f;
#pragma unroll
    for (int k = 0; k < 8; ++k) total += accv[k];
    total *= 0.5f;

#pragma unroll
    for (int off = 16; off > 0; off >>= 1) total += __shfl_down(total, off, 32);

    if (lane == 0) {
        atomicAdd(&ws[0], total);        // global hinge sum
        atomicAdd(&ws[1], nrel * nirr);  // global pair count
    }
}

__global__ void rag_finalize(const float* __restrict__ ws, float* __restrict__ out) {
    out[0] = ws[0] / (ws[1] + 1e-6f);
}

extern "C" void kernel_launch(void* const* d_in, const int* in_sizes, int n_in,
                              void* d_out, int out_size, void* d_ws, size_t ws_size,
                              hipStream_t stream) {
    const float* scores    = (const float*)d_in[0];
    const int*   relevance = (const int*)d_in[1];
    float* ws  = (float*)d_ws;
    float* out = (float*)d_out;

    rag_zero_ws<<<1, 32, 0, stream>>>(ws);
    rag_loss_kernel<<<RAG_BATCH / WAVES_PER_BLOCK, WAVES_PER_BLOCK * 32, 0, stream>>>(
        scores, relevance, ws);
    rag_finalize<<<1, 1, 0, stream>>>(ws, out);
}